// AttentionDecoder_46952582480397
// MI455X (gfx1250) — compile-verified
//
#include <hip/hip_runtime.h>
#include <hip/hip_bf16.h>

#define BSZ    128
#define NNODES 2000
#define DIM    256
#define NH     8
#define HDIM   32
#define CCTX   64
#define CNODE  8
#define BLOCK  512

typedef __attribute__((ext_vector_type(2))) float v2f;
typedef __attribute__((ext_vector_type(8))) float v8f;

// D(16x16 f32) = A(16x4 f32) x B(4x16 f32) + C
__device__ __forceinline__ v8f wmma4(v2f a, v2f b, v8f c) {
  return __builtin_amdgcn_wmma_f32_16x16x4_f32(false, a, false, b, (short)0, c, false, false);
}

__global__ __launch_bounds__(BLOCK) void attn_decoder_kernel(
    const float* __restrict__ shelf, const float* __restrict__ enc,
    const float* __restrict__ ctx,   const float* __restrict__ nf,
    const float* __restrict__ Wk,    const float* __restrict__ Wv,
    const float* __restrict__ Wk2,   const float* __restrict__ Wq,
    const float* __restrict__ Wout,  const float* __restrict__ Wc,
    const float* __restrict__ Wg,    const float* __restrict__ Wn,
    const int* __restrict__ cur_node, const unsigned char* __restrict__ mask,
    float* __restrict__ out)
{
  // Row 8 of sComp/sQt/sR is an always-zero pad row so WMMA A-operand rows
  // M=8..15 need no per-iteration masking (base-pointer select only).
  __shared__ float sComp[NH + 1][NNODES];    // compat -> attnExp; row 0 reused as logits
  __shared__ float sQt[(NH + 1) * DIM];      // q-tilde (pre-scaled by 1/sqrt(HD))
  __shared__ float sR[(NH + 1) * CNODE];     // query folded through Wn (k_u part)
  __shared__ float sS[NH * DIM];             // unnormalized attn-weighted enc sums
  __shared__ float sRed[BLOCK];
  __shared__ float sMean[DIM], sX1[DIM], sX2[DIM], sGE[DIM], sQ[DIM];
  __shared__ float sHeads[DIM], sGl[DIM], sGt[DIM];
  __shared__ float sT[NH * CNODE], sGn[CNODE];
  __shared__ float sHmax[NH], sHsum[NH];

  const int b    = blockIdx.x;
  const int tid  = threadIdx.x;
  const int lane = tid & 31;
  const int wave = tid >> 5;
  const float* encB = enc + (size_t)b * NNODES * DIM;
  const float* nfB  = nf  + (size_t)b * NNODES * CNODE;
  const unsigned char* maskB = mask + (size_t)b * NNODES;
  float* outB = out + (size_t)b * NNODES;
  const float RS32  = 0.17677669529663687f;  // 1/sqrt(HD)
  const float INV16 = 0.0625f;               // 1/sqrt(D)

  // ---- Phase A: mean over nodes (coalesced: 256 threads span one row) ----
  {
    float acc = 0.f;
    const int d = tid & (DIM - 1);
    const int nbeg = (tid >> 8) * (NNODES / 2);
    #pragma unroll 4
    for (int n = nbeg; n < nbeg + NNODES / 2; ++n) acc += encB[(size_t)n * DIM + d];
    sRed[tid] = acc;
    __syncthreads();
    if (tid < DIM) sMean[tid] = (sRed[tid] + sRed[tid + DIM]) * (1.0f / NNODES);
    __syncthreads();
  }

  // ---- Phase B: per-batch query math + zero the pad rows ----
  const int cn = cur_node[b];
  for (int n = tid; n < NNODES; n += BLOCK) sComp[NH][n] = 0.f;   // zero pad row for Phase E
  if (tid < DIM) {
    sQt[NH * DIM + tid] = 0.f;                                    // zero pad row for Phase C
    float cx = 0.f;
    for (int c = 0; c < CCTX; ++c) cx += Wc[(size_t)tid * CCTX + c] * ctx[(size_t)b * CCTX + c];
    sX2[tid] = cx;                                    // context projection
    float g = 0.f;
    for (int d = 0; d < DIM; ++d) g += Wg[(size_t)tid * DIM + d] * sMean[d];
    sGE[tid] = g;                                     // graph embed
  } else {
    const int d = tid - DIM;
    if (d < CNODE) sR[NH * CNODE + d] = 0.f;                      // zero pad row
    sX1[d] = shelf[((size_t)b * NNODES + cn) * DIM + d];          // current node embedding
  }
  __syncthreads();
  if (tid < DIM) {                                    // glimpse_Q = X @ Wq^T + graph_embed
    float q = sGE[tid];
    const float* wq = Wq + (size_t)tid * (2 * DIM);
    for (int j = 0; j < DIM; ++j) q += wq[j] * sX1[j];
    for (int j = 0; j < DIM; ++j) q += wq[DIM + j] * sX2[j];
    sQ[tid] = q;
  }
  __syncthreads();
  // q-tilde[h,d] = sum_e Q[32h+e] * Wk[32h+e, d]  (query folded through Wk)
  for (int i = 0; i < 4; ++i) {
    const int o = tid + BLOCK * i;                    // 0..2047, consecutive d per tid
    const int h = o >> 8, d = o & (DIM - 1);
    float s = 0.f;
    for (int e = 0; e < HDIM; ++e) s += sQ[h * HDIM + e] * Wk[(size_t)(h * HDIM + e) * DIM + d];
    sQt[o] = s * RS32;
  }
  if (tid < NH * CNODE) {                             // r[h,c]: query folded through Wn (k_u part)
    const int h = tid >> 3, c = tid & 7;
    float s = 0.f;
    for (int e = 0; e < HDIM; ++e) s += sQ[h * HDIM + e] * Wn[(size_t)(h * HDIM + e) * CNODE + c];
    sR[tid] = s * RS32;
  }
  __syncthreads();

  // ---- Phase C: compat[h,n] via WMMA: C[16h x 16n] += A(qt 16x4) x B(enc^T 4x16) ----
  {
    const int mrow  = lane & 15, khalf = lane >> 4;
    const int arow  = (mrow < 8) ? mrow : 8;          // pad row for M=8..15 (loop-invariant)
    const float* aQ = &sQt[arow * DIM + 2 * khalf];
    const float* aR = &sR[arow * CNODE + 2 * khalf];
    for (int tile = wave; tile < NNODES / 16; tile += 16) {
      const int nb = tile * 16;
      const float* brow = encB + (size_t)(nb + mrow) * DIM + 2 * khalf;
      const float* nrow = nfB  + (size_t)(nb + mrow) * CNODE + 2 * khalf;
      v8f c = {};
      #pragma unroll 4
      for (int k0 = 0; k0 < DIM; k0 += 4) {
        v2f a = {aQ[k0], aQ[k0 + 1]};
        v2f bb = {brow[k0], brow[k0 + 1]};
        c = wmma4(a, bb, c);
      }
      #pragma unroll
      for (int c0 = 0; c0 < CNODE; c0 += 4) {         // dynamic node-feature term
        v2f a = {aR[c0], aR[c0 + 1]};
        v2f bb = {nrow[c0], nrow[c0 + 1]};
        c = wmma4(a, bb, c);
      }
      if (lane < 16) {                                // C: VGPR j = row M=j (head j), lanes 0..15 = cols
        #pragma unroll
        for (int j = 0; j < 8; ++j) sComp[j][nb + lane] = c[j];
      }
    }
  }
  __syncthreads();

  // ---- Phase D: per-head softmax stats; exp in place ----
  {
    const int g = tid >> 6, l = tid & 63;
    float mx = -__builtin_inff();
    for (int n = l; n < NNODES; n += 64) mx = fmaxf(mx, sComp[g][n]);
    sRed[tid] = mx; __syncthreads();
    for (int s = 32; s >= 1; s >>= 1) { if (l < s) sRed[tid] = fmaxf(sRed[tid], sRed[tid + s]); __syncthreads(); }
    if (l == 0) sHmax[g] = sRed[tid];
    __syncthreads();
    const float hm = sHmax[g];
    float sm = 0.f;
    for (int n = l; n < NNODES; n += 64) { float e = __expf(sComp[g][n] - hm); sComp[g][n] = e; sm += e; }
    sRed[tid] = sm; __syncthreads();
    for (int s = 32; s >= 1; s >>= 1) { if (l < s) sRed[tid] += sRed[tid + s]; __syncthreads(); }
    if (l == 0) sHsum[g] = sRed[tid];
    __syncthreads();
  }

  // ---- Phase E: S[h,d] = sum_n attnExp[h,n] enc[n,d] via WMMA (wave w -> dim chunk w*16) ----
  {
    const int mrow  = lane & 15, khalf = lane >> 4;
    const int arow  = (mrow < 8) ? mrow : 8;          // pad row (zeroed in Phase B)
    const float* aC = &sComp[arow][2 * khalf];
    const int d0 = wave * 16;
    const int dcol = d0 + mrow;
    v8f c = {};
    #pragma unroll 2
    for (int n0 = 0; n0 < NNODES; n0 += 4) {
      v2f a = {aC[n0], aC[n0 + 1]};
      const size_t kk = (size_t)(n0 + 2 * khalf);
      v2f bb = {encB[kk * DIM + dcol], encB[(kk + 1) * DIM + dcol]};
      c = wmma4(a, bb, c);
    }
    if (lane < 16) {
      #pragma unroll
      for (int j = 0; j < 8; ++j) sS[j * DIM + d0 + lane] = c[j];
    }
  }
  __syncthreads();

  // ---- t[h,c] = sum_n attnExp[h,n] * nf[n,c] ----
  {
    const int o = tid & 63, p = tid >> 6;
    const int h = o >> 3, cc = o & 7;
    float tp = 0.f;
    for (int n = p; n < NNODES; n += 8) tp += sComp[h][n] * nfB[(size_t)n * CNODE + cc];
    sRed[tid] = tp; __syncthreads();
    if (tid < 64) { float s = 0.f; for (int q = 0; q < 8; ++q) s += sRed[tid + 64 * q]; sT[tid] = s; }
    __syncthreads();
  }

  // ---- Phase F: heads -> glimpse -> folded logit keys ----
  if (tid < DIM) {                                    // heads_flat[32h+e] (normalized)
    const int h = tid >> 5;
    float v = 0.f;
    const float* wr = Wv + (size_t)tid * DIM;
    for (int d = 0; d < DIM; ++d) v += wr[d] * sS[h * DIM + d];
    for (int c = 0; c < CNODE; ++c) v += Wn[(size_t)(DIM + tid) * CNODE + c] * sT[h * CNODE + c];
    sHeads[tid] = v / sHsum[h];
  }
  __syncthreads();
  if (tid < DIM) {                                    // glimpse = heads @ Wout^T
    float g = 0.f;
    const float* wr = Wout + (size_t)tid * DIM;
    for (int k = 0; k < DIM; ++k) g += wr[k] * sHeads[k];
    sGl[tid] = g;
  }
  __syncthreads();
  if (tid < DIM) {                                    // g-tilde = Wk2^T glimpse / sqrt(D)
    float g = 0.f;
    for (int dp = 0; dp < DIM; ++dp) g += sGl[dp] * Wk2[(size_t)dp * DIM + tid];
    sGt[tid] = g * INV16;
  } else if (tid < DIM + CNODE) {                     // gn = Wn_l^T glimpse / sqrt(D)
    const int cc = tid - DIM;
    float g = 0.f;
    for (int dp = 0; dp < DIM; ++dp) g += sGl[dp] * Wn[(size_t)(2 * DIM + dp) * CNODE + cc];
    sGn[cc] = g * INV16;
  }
  __syncthreads();

  // ---- Phase G: logits (wave per node, coalesced), masked softmax ----
  float* sL = &sComp[0][0];
  for (int n = wave; n < NNODES; n += 16) {
    const float* er = encB + (size_t)n * DIM;
    float acc = 0.f;
    #pragma unroll
    for (int k = 0; k < 8; ++k) acc += sGt[lane + 32 * k] * er[lane + 32 * k];
    for (int off = 16; off > 0; off >>= 1) acc += __shfl_xor(acc, off, 32);
    if (lane == 0) {
      float nd = 0.f;
      const float* nr = nfB + (size_t)n * CNODE;
      for (int c = 0; c < CNODE; ++c) nd += sGn[c] * nr[c];
      float l = 10.f * tanhf(acc + nd);
      if (maskB[n]) l = -__builtin_inff();
      sL[n] = l;
    }
  }
  __syncthreads();
  float m = -__builtin_inff();
  for (int n = tid; n < NNODES; n += BLOCK) m = fmaxf(m, sL[n]);
  sRed[tid] = m; __syncthreads();
  for (int s = 256; s >= 1; s >>= 1) { if (tid < s) sRed[tid] = fmaxf(sRed[tid], sRed[tid + s]); __syncthreads(); }
  const float gmax = sRed[0];
  __syncthreads();
  float ps = 0.f;
  for (int n = tid; n < NNODES; n += BLOCK) { float e = __expf(sL[n] - gmax); sL[n] = e; ps += e; }
  sRed[tid] = ps; __syncthreads();
  for (int s = 256; s >= 1; s >>= 1) { if (tid < s) sRed[tid] += sRed[tid + s]; __syncthreads(); }
  const float inv = 1.0f / sRed[0];
  for (int n = tid; n < NNODES; n += BLOCK) outB[n] = sL[n] * inv;
}

extern "C" void kernel_launch(void* const* d_in, const int* in_sizes, int n_in,
                              void* d_out, int out_size, void* d_ws, size_t ws_size,
                              hipStream_t stream) {
  dim3 grid(BSZ), block(BLOCK);
  attn_decoder_kernel<<<grid, block, 0, stream>>>(
      (const float*)d_in[0],  (const float*)d_in[1],  (const float*)d_in[2],
      (const float*)d_in[3],  (const float*)d_in[4],  (const float*)d_in[5],
      (const float*)d_in[6],  (const float*)d_in[7],  (const float*)d_in[8],
      (const float*)d_in[9],  (const float*)d_in[10], (const float*)d_in[11],
      (const int*)d_in[12],   (const unsigned char*)d_in[13],
      (float*)d_out);
}